// HGT_7851200217520
// MI455X (gfx1250) — compile-verified
//
#include <hip/hip_runtime.h>
#include <hip/hip_bf16.h>
#include <math.h>

typedef __attribute__((ext_vector_type(2))) float v2f;
typedef __attribute__((ext_vector_type(8))) float v8f;
typedef __attribute__((vector_size(16))) int i32x4;

#define N_NODES 20000
#define E_EDGES 320000
#define INDIM   256

// ---------------- async-to-LDS staging (gfx1250), with sync fallback --------
#if __has_builtin(__builtin_amdgcn_global_load_async_to_lds_b128) && \
    __has_builtin(__builtin_amdgcn_global_load_async_to_lds_b32)  && \
    __has_builtin(__builtin_amdgcn_s_wait_asynccnt)
#define HAVE_ASYNC_LDS 1
#else
#define HAVE_ASYNC_LDS 0
#endif

__device__ __forceinline__ void stage_b128(const float* gsrc, float* ldst) {
#if HAVE_ASYNC_LDS
    __builtin_amdgcn_global_load_async_to_lds_b128(
        (__attribute__((address_space(1))) i32x4*)gsrc,
        (__attribute__((address_space(3))) i32x4*)ldst, 0, 0);
#else
    *(float4*)ldst = *(const float4*)gsrc;
#endif
}
__device__ __forceinline__ void stage_b32(const float* gsrc, float* ldst) {
#if HAVE_ASYNC_LDS
    __builtin_amdgcn_global_load_async_to_lds_b32(
        (__attribute__((address_space(1))) int*)gsrc,
        (__attribute__((address_space(3))) int*)ldst, 0, 0);
#else
    *ldst = *gsrc;
#endif
}
__device__ __forceinline__ void stage_wait() {
#if HAVE_ASYNC_LDS
    __builtin_amdgcn_s_wait_asynccnt(0);
#endif
    __syncthreads();
}

// ---------- order-preserving float<->uint map for atomic max ----------------
__device__ __forceinline__ unsigned int fmap(float x) {
    unsigned int b = __float_as_uint(x);
    return (b & 0x80000000u) ? ~b : (b | 0x80000000u);
}
__device__ __forceinline__ float funmap(unsigned int m) {
    unsigned int b = (m & 0x80000000u) ? (m & 0x7FFFFFFFu) : ~m;
    return __uint_as_float(b);
}

// ---------------------------------------------------------------------------
// GEMM: C[N x 256] = A[N x 256] @ W[256 x 256], fp32 WMMA 16x16x4.
// 32-row tile per block (grid = N/32); each wave keeps two accumulators so
// every B-fragment load feeds TWO wmmas. A tile staged in LDS via async-to-LDS
// (stride 260 => conflict-free 16x4 fragment reads).
// SKIP: C = C*sigmoid(*skip_s) + Xskip*(1-sigmoid(*skip_s)).
// ---------------------------------------------------------------------------
template <int SKIP>
__launch_bounds__(256)
__global__ void gemm_wmma(const float* __restrict__ A,
                          const float* __restrict__ W,
                          float* __restrict__ C,
                          const float* __restrict__ Xskip,
                          const float* __restrict__ skip_s)
{
    __shared__ float ldsA[32 * 260];
    const int m0  = blockIdx.x * 32;
    const int tid = threadIdx.x;

    // stage 32x256 A tile: 32 rows * 64 float4, 8 per thread
    for (int i = tid; i < 32 * 64; i += 256) {
        const int r  = i >> 6;
        const int c4 = i & 63;
        stage_b128(A + (size_t)(m0 + r) * INDIM + c4 * 4, &ldsA[r * 260 + c4 * 4]);
    }
    stage_wait();

    const int lane  = tid & 31;
    const int wave  = tid >> 5;
    const int row   = lane & 15;          // A-fragment row (M)
    const int khalf = (lane >> 4) * 2;    // lanes 16-31 hold K+2, K+3

    float alpha = 1.0f, beta = 0.0f;
    if (SKIP) {
        const float s = *skip_s;
        alpha = 1.0f / (1.0f + __expf(-s));
        beta  = 1.0f - alpha;
    }

    for (int t = 0; t < 2; ++t) {
        const int n0  = (wave * 2 + t) * 16;
        const int col = n0 + (lane & 15);
        v8f c0 = {}, c1 = {};
#pragma unroll 4
        for (int k = 0; k < 256; k += 4) {
            const int kk = k + khalf;
            v2f a0, a1, b;
            a0.x = ldsA[row * 260 + kk];
            a0.y = ldsA[row * 260 + kk + 1];
            a1.x = ldsA[(row + 16) * 260 + kk];
            a1.y = ldsA[(row + 16) * 260 + kk + 1];
            b.x  = W[(size_t)kk * 256 + col];
            b.y  = W[(size_t)(kk + 1) * 256 + col];
            c0 = __builtin_amdgcn_wmma_f32_16x16x4_f32(
                     false, a0, false, b, (short)0, c0, false, false);
            c1 = __builtin_amdgcn_wmma_f32_16x16x4_f32(
                     false, a1, false, b, (short)0, c1, false, false);
        }
        // C/D layout: VGPR r -> M = r (lanes 0-15) / r+8 (lanes 16-31)
        const int rbase = m0 + ((lane >> 4) << 3);
#pragma unroll
        for (int r = 0; r < 8; ++r) {
            const size_t i0 = (size_t)(rbase + r) * 256 + col;
            const size_t i1 = (size_t)(rbase + 16 + r) * 256 + col;
            float v0 = c0[r], v1 = c1[r];
            if (SKIP) {
                v0 = v0 * alpha + Xskip[i0] * beta;
                v1 = v1 * alpha + Xskip[i1] * beta;
            }
            C[i0] = v0;
            C[i1] = v1;
        }
    }
}

// ---------------------------------------------------------------------------
// Edge pass 1: a[e,h] = (q_e . (k_e @ Watt[h,et])) * pri[h,et] / sqrt(D)
// plus segment-max via mapped-uint atomicMax. Wave per edge, lane per head.
// Watt (H*2*64 floats for this layer) staged in LDS, stride 130/head.
// ---------------------------------------------------------------------------
__launch_bounds__(256)
__global__ void edge_attn(const float* __restrict__ K,
                          const float* __restrict__ Q,
                          const float* __restrict__ Wl,     // H*2*64
                          const float* __restrict__ pril,   // H*2
                          const int* __restrict__ src,
                          const int* __restrict__ dst,
                          const int* __restrict__ etype,
                          float* __restrict__ aout,         // E*H
                          unsigned int* __restrict__ amax_m)// N*H (mapped)
{
    __shared__ float ldsW[32 * 130];
    for (int i = threadIdx.x; i < 32 * 128; i += 256) {
        const int h = i >> 7, off = i & 127;
        stage_b32(Wl + i, &ldsW[h * 130 + off]);
    }
    stage_wait();

    const int lane = threadIdx.x & 31;   // head
    const int wave = threadIdx.x >> 5;
    const float inv_sqrt_d = 0.35355339059327373f;

    for (int e = blockIdx.x * 8 + wave; e < E_EDGES; e += gridDim.x * 8) {
        const int s  = src[e];
        const int d  = dst[e];
        const int et = etype[e];
        const float4* kp = (const float4*)(K + (size_t)s * 256 + lane * 8);
        const float4* qp = (const float4*)(Q + (size_t)d * 256 + lane * 8);
        const float4 k0 = kp[0], k1 = kp[1];
        const float4 q0 = qp[0], q1 = qp[1];
        const float kv[8] = {k0.x, k0.y, k0.z, k0.w, k1.x, k1.y, k1.z, k1.w};
        const float qv[8] = {q0.x, q0.y, q0.z, q0.w, q1.x, q1.y, q1.z, q1.w};
        const float* w = &ldsW[lane * 130 + et * 64];

        float acc = 0.0f;
#pragma unroll
        for (int f = 0; f < 8; ++f) {
            float kw = 0.0f;
#pragma unroll
            for (int dd = 0; dd < 8; ++dd)
                kw = __builtin_fmaf(kv[dd], w[dd * 8 + f], kw);
            acc = __builtin_fmaf(kw, qv[f], acc);
        }
        acc *= pril[lane * 2 + et] * inv_sqrt_d;
        aout[(size_t)e * 32 + lane] = acc;
        atomicMax(&amax_m[(size_t)d * 32 + lane], fmap(acc));
    }
}

// ---------------------------------------------------------------------------
// Edge pass 2: denom[dst,h] += exp(a - amax[dst,h]).  Thread per (e,h).
// ---------------------------------------------------------------------------
__launch_bounds__(256)
__global__ void edge_denom(const float* __restrict__ a,
                           const int* __restrict__ dst,
                           const unsigned int* __restrict__ amax_m,
                           float* __restrict__ denom)
{
    const size_t i = (size_t)blockIdx.x * blockDim.x + threadIdx.x;
    if (i >= (size_t)E_EDGES * 32) return;
    const int e = (int)(i >> 5);
    const int h = (int)(i & 31);
    const int d = dst[e];
    const float amax = funmap(amax_m[(size_t)d * 32 + h]);
    atomicAdd(&denom[(size_t)d * 32 + h], __expf(a[i] - amax));
}

// ---------------------------------------------------------------------------
// Edge pass 3: m = v_e @ Wmsg[h,et]; att = exp(a-amax)/denom;
// hagg[dst, h*8+f] += m[f]*att.  Wave per edge, lane per head.
// ---------------------------------------------------------------------------
__launch_bounds__(256)
__global__ void edge_msg(const float* __restrict__ V,
                         const float* __restrict__ Wl,     // Wmsg layer: H*2*64
                         const float* __restrict__ a,
                         const unsigned int* __restrict__ amax_m,
                         const float* __restrict__ denom,
                         const int* __restrict__ src,
                         const int* __restrict__ dst,
                         const int* __restrict__ etype,
                         float* __restrict__ hagg)
{
    __shared__ float ldsW[32 * 130];
    for (int i = threadIdx.x; i < 32 * 128; i += 256) {
        const int h = i >> 7, off = i & 127;
        stage_b32(Wl + i, &ldsW[h * 130 + off]);
    }
    stage_wait();

    const int lane = threadIdx.x & 31;
    const int wave = threadIdx.x >> 5;

    for (int e = blockIdx.x * 8 + wave; e < E_EDGES; e += gridDim.x * 8) {
        const int s  = src[e];
        const int d  = dst[e];
        const int et = etype[e];
        const float av   = a[(size_t)e * 32 + lane];
        const float amax = funmap(amax_m[(size_t)d * 32 + lane]);
        const float att  = __expf(av - amax) / denom[(size_t)d * 32 + lane];

        const float4* vp = (const float4*)(V + (size_t)s * 256 + lane * 8);
        const float4 v0 = vp[0], v1 = vp[1];
        const float vv[8] = {v0.x, v0.y, v0.z, v0.w, v1.x, v1.y, v1.z, v1.w};
        const float* w = &ldsW[lane * 130 + et * 64];

        float* hrow = &hagg[(size_t)d * 256 + lane * 8];
#pragma unroll
        for (int f = 0; f < 8; ++f) {
            float m = 0.0f;
#pragma unroll
            for (int dd = 0; dd < 8; ++dd)
                m = __builtin_fmaf(vv[dd], w[dd * 8 + f], m);
            atomicAdd(&hrow[f], m * att);
        }
    }
}

// ---------------------------------------------------------------------------
extern "C" void kernel_launch(void* const* d_in, const int* in_sizes, int n_in,
                              void* d_out, int out_size, void* d_ws, size_t ws_size,
                              hipStream_t stream) {
    const float* x     = (const float*)d_in[0];
    const float* Wk    = (const float*)d_in[1];   // (2,256,256)
    const float* Wq    = (const float*)d_in[2];
    const float* Wv    = (const float*)d_in[3];
    const float* Wa    = (const float*)d_in[4];   // (2,256,256)
    const float* Watt  = (const float*)d_in[5];   // (2,32,2,8,8)
    const float* Wmsg  = (const float*)d_in[6];
    const float* pri   = (const float*)d_in[7];   // (2,32,2)
    const float* skip  = (const float*)d_in[8];   // (2,)
    const int*   src   = (const int*)d_in[9];
    const int*   dst   = (const int*)d_in[10];
    const int*   etype = (const int*)d_in[11];
    float* out = (float*)d_out;

    // workspace layout (floats)
    float* ws = (float*)d_ws;
    const size_t nd = (size_t)N_NODES * 256;
    float* Kb    = ws;                 // N*256
    float* Qb    = Kb + nd;            // N*256
    float* Vb    = Qb + nd;            // N*256
    float* hbuf  = Vb + nd;            // N*256 (layer-0 output)
    float* hagg  = hbuf + nd;          // N*256
    float* ab    = hagg + nd;          // E*32
    unsigned int* amax = (unsigned int*)(ab + (size_t)E_EDGES * 32); // N*32
    float* denom = (float*)(amax + (size_t)N_NODES * 32);            // N*32

    const int gemm_grid = N_NODES / 32;               // 625
    const int edge_grid = 2048;                       // wave-per-edge, strided
    const int eh_grid   = (int)(((size_t)E_EDGES * 32 + 255) / 256);

    for (int l = 0; l < 2; ++l) {
        const float* xin  = (l == 0) ? x : hbuf;
        float*       xout = (l == 1) ? out : hbuf;
        const size_t wOff = (size_t)l * 256 * 256;
        const size_t rOff = (size_t)l * 32 * 2 * 64;
        const size_t pOff = (size_t)l * 32 * 2;

        gemm_wmma<0><<<gemm_grid, 256, 0, stream>>>(xin, Wk + wOff, Kb, nullptr, nullptr);
        gemm_wmma<0><<<gemm_grid, 256, 0, stream>>>(xin, Wq + wOff, Qb, nullptr, nullptr);
        gemm_wmma<0><<<gemm_grid, 256, 0, stream>>>(xin, Wv + wOff, Vb, nullptr, nullptr);

        (void)hipMemsetAsync(amax,  0, (size_t)N_NODES * 32 * 4, stream);
        (void)hipMemsetAsync(denom, 0, (size_t)N_NODES * 32 * 4, stream);
        (void)hipMemsetAsync(hagg,  0, nd * 4, stream);

        edge_attn<<<edge_grid, 256, 0, stream>>>(Kb, Qb, Watt + rOff, pri + pOff,
                                                 src, dst, etype, ab, amax);
        edge_denom<<<eh_grid, 256, 0, stream>>>(ab, dst, amax, denom);
        edge_msg<<<edge_grid, 256, 0, stream>>>(Vb, Wmsg + rOff, ab, amax, denom,
                                                src, dst, etype, hagg);

        gemm_wmma<1><<<gemm_grid, 256, 0, stream>>>(hagg, Wa + wOff, xout, xin, skip + l);
    }
}